// BPGNN_38036230373427
// MI455X (gfx1250) — compile-verified
//
#include <hip/hip_runtime.h>
#include <hip/hip_bf16.h>
#include <math.h>

// ---------------------------------------------------------------------------
// BP-GNN forward on gfx1250 (MI455X).
//   N=100000 nodes, E=3.2M directed edges (pairs e <-> e+E2), C=8, K=5.
//   Stage 1: log_b0 = log_softmax(x @ W + b)  -> V_WMMA_F32_16X16X4_F32
//            (W pre-padded to Kx16, pair-interleaved, so each B fragment is
//             one unconditional global_load_b64 -> no EXEC juggling in loop)
//   Stage 2: K BP sweeps: per-edge-pair logsumexp message update (VALU/trans
//            bound, random gathers -> global b128 loads + f32 atomics).
// ---------------------------------------------------------------------------

typedef float v2f __attribute__((ext_vector_type(2)));
typedef float v8f __attribute__((ext_vector_type(8)));

#define CDIM 8
#define KDIM 128
#define NEG_LOG_C (-2.0794415416798357f) /* -log(8) */

// ---------------- logH build (8x8, diag=0, symmetric log_sigmoid) ----------
__global__ void k_logH(const float* __restrict__ param, float* __restrict__ H) {
  int t = threadIdx.x;            // 64 threads
  int i = t >> 3, j = t & 7;
  float v = 0.0f;
  if (i != j) {
    int mx = i > j ? i : j, mn = i > j ? j : i;
    float p = param[mx * (mx + 1) / 2 + mn];
    // stable log_sigmoid(p) = min(p,0) - log1p(exp(-|p|))
    v = fminf(p, 0.0f) - log1pf(__expf(-fabsf(p)));
  }
  H[i * 8 + j] = v;
}

// ---------------- W pre-pad: Wpad[kpair][n][half] = W[2*kpair+half][n] ------
// (zero for n >= 8).  One b64 load then yields B VGPR0/1 for a WMMA step.
__global__ void k_wpad(const float* __restrict__ W, float* __restrict__ Wpad) {
  int t = blockIdx.x * blockDim.x + threadIdx.x;   // 0 .. (KDIM/2)*16*2 - 1
  if (t >= (KDIM / 2) * 32) return;
  int kpair = t >> 5;
  int rem   = t & 31;
  int n     = rem >> 1;
  int half  = rem & 1;
  Wpad[t] = (n < CDIM) ? W[(2 * kpair + half) * CDIM + n] : 0.0f;
}

// ---------------- WMMA GEMM + log_softmax ----------------------------------
// One wave computes one 16-row tile of logits = x[16xK] @ Wpad[Kx16] via
// 32 chained V_WMMA_F32_16X16X4_F32, then does a row softmax through LDS.
__global__ __launch_bounds__(256) void k_gemm_logsoftmax(
    const float* __restrict__ x, const float* __restrict__ Wpad,
    const float* __restrict__ bias, float* __restrict__ log_b0,
    float* __restrict__ log_b, int N) {
  __shared__ float lds[8][16 * 9];
  const int lane = threadIdx.x & 31;
  const int wave = threadIdx.x >> 5;
  const int m15  = lane & 15;       // M (for A/C) or N (for B/C)
  const int half = lane >> 4;       // K-half selector
  const long tile = (long)blockIdx.x * 8 + wave;
  const long row0 = tile * 16;
  const bool tile_valid = (row0 < (long)N);  // wave-uniform (EXEC stays all-1)

  if (tile_valid) {
    long arow = row0 + m15;
    if (arow >= N) arow = N - 1;           // clamp tail-row loads
    const float* aptr = x + arow * (long)KDIM + half * 2;
    const float* bptr = Wpad + m15 * 2 + half * 32;
    v8f c = {};
#pragma unroll
    for (int k0 = 0; k0 < KDIM; k0 += 4) {
      // A (16x4 f32): VGPR j, lane l -> M=l&15, K=k0+(l>>4)*2+j
      v2f a = *(const v2f*)(aptr + k0);
      // B (4x16 f32): VGPR j, lane l -> K=k0+(l>>4)*2+j, N=l&15 (pad -> 0)
      v2f bb = *(const v2f*)(bptr + (k0 >> 1) * 32);
      c = __builtin_amdgcn_wmma_f32_16x16x4_f32(
          /*neg_a=*/false, a, /*neg_b=*/false, bb,
          /*c_mod=*/(short)0, c, /*reuse_a=*/false, /*reuse_b=*/false);
    }
    // C/D layout: VGPR r, lane l -> M = r + (l>>4)*8, N = l&15
    if (m15 < CDIM) {
      float bj = bias[m15];
#pragma unroll
      for (int r = 0; r < 8; ++r) {
        int M = r + half * 8;
        lds[wave][M * 9 + m15] = c[r] + bj;
      }
    }
  }
  __syncthreads();
  if (tile_valid && lane < 16) {
    long row = row0 + lane;
    if (row < N) {
      float v[CDIM], mx = -INFINITY;
#pragma unroll
      for (int j = 0; j < CDIM; ++j) {
        v[j] = lds[wave][lane * 9 + j];
        mx = fmaxf(mx, v[j]);
      }
      float s = 0.0f;
#pragma unroll
      for (int j = 0; j < CDIM; ++j) s += __expf(v[j] - mx);
      float lse = mx + __logf(s);
      long base = row * CDIM;
#pragma unroll
      for (int j = 0; j < CDIM; ++j) {
        float o = v[j] - lse;
        log_b0[base + j] = o;   // saved for every iteration
        log_b[base + j]  = o;   // initial beliefs (d_out)
      }
    }
  }
}

// ---------------- state init (ws is poisoned by harness) -------------------
__global__ void k_init(float* __restrict__ log_msg, float* __restrict__ agg,
                       size_t n_msg, size_t n_agg) {
  size_t idx = (size_t)blockIdx.x * blockDim.x + threadIdx.x;
  if (idx < n_msg) log_msg[idx] = NEG_LOG_C;
  if (idx < n_agg) agg[idx] = 0.0f;
}

// ---------------- per-edge message math ------------------------------------
__device__ __forceinline__ void compute_msg(const float* __restrict__ a,
                                            float w,
                                            const float* __restrict__ H,
                                            float* __restrict__ out) {
  float mx2 = -INFINITY;
#pragma unroll
  for (int j = 0; j < CDIM; ++j) {
    float m = -INFINITY;
#pragma unroll
    for (int i = 0; i < CDIM; ++i) m = fmaxf(m, a[i] + w * H[i * 8 + j]);
    float s = 0.0f;
#pragma unroll
    for (int i = 0; i < CDIM; ++i) s += __expf(a[i] + w * H[i * 8 + j] - m);
    out[j] = m + __logf(s);
    mx2 = fmaxf(mx2, out[j]);
  }
  float s2 = 0.0f;
#pragma unroll
  for (int j = 0; j < CDIM; ++j) s2 += __expf(out[j] - mx2);
  float lse = mx2 + __logf(s2);
#pragma unroll
  for (int j = 0; j < CDIM; ++j) out[j] -= lse;
}

// One thread per edge PAIR (e, e+E2): rv[e]=e+E2, rv[e+E2]=e, so handling the
// pair in one thread removes the need to double-buffer log_msg.
__global__ __launch_bounds__(256) void k_edge(
    const int* __restrict__ src, const int* __restrict__ dst,
    const float* __restrict__ ew, const float* __restrict__ log_b,
    float* __restrict__ log_msg, float* __restrict__ agg,
    const float* __restrict__ Hg, int E2) {
  __shared__ float H[64];
  if (threadIdx.x < 64) H[threadIdx.x] = Hg[threadIdx.x];
  __syncthreads();

  int i = blockIdx.x * blockDim.x + threadIdx.x;
  if (i >= E2) return;
  const int e1 = i, e2 = i + E2;
  const int u1 = src[e1], v1 = dst[e1];
  const int u2 = src[e2], v2 = dst[e2];
  const float w1 = ew[e1], w2 = ew[e2];

  // 128-bit loads: rows are 32B and 32B-aligned.
  const float4* mr1 = (const float4*)(log_msg + (size_t)e1 * CDIM);
  const float4* mr2 = (const float4*)(log_msg + (size_t)e2 * CDIM);
  const float4* xr1 = (const float4*)(log_b + (size_t)u1 * CDIM);
  const float4* xr2 = (const float4*)(log_b + (size_t)u2 * CDIM);
  float4 m1a = mr1[0], m1b = mr1[1];
  float4 m2a = mr2[0], m2b = mr2[1];
  float4 x1a = xr1[0], x1b = xr1[1];
  float4 x2a = xr2[0], x2b = xr2[1];

  float a1[CDIM], a2[CDIM];
  a1[0] = x1a.x - m2a.x; a1[1] = x1a.y - m2a.y;
  a1[2] = x1a.z - m2a.z; a1[3] = x1a.w - m2a.w;
  a1[4] = x1b.x - m2b.x; a1[5] = x1b.y - m2b.y;
  a1[6] = x1b.z - m2b.z; a1[7] = x1b.w - m2b.w;
  a2[0] = x2a.x - m1a.x; a2[1] = x2a.y - m1a.y;
  a2[2] = x2a.z - m1a.z; a2[3] = x2a.w - m1a.w;
  a2[4] = x2b.x - m1b.x; a2[5] = x2b.y - m1b.y;
  a2[6] = x2b.z - m1b.z; a2[7] = x2b.w - m1b.w;

  float n1[CDIM], n2[CDIM];
  compute_msg(a1, w1, H, n1);
  compute_msg(a2, w2, H, n2);

  float4* wr1 = (float4*)(log_msg + (size_t)e1 * CDIM);
  float4* wr2 = (float4*)(log_msg + (size_t)e2 * CDIM);
  wr1[0] = make_float4(n1[0], n1[1], n1[2], n1[3]);
  wr1[1] = make_float4(n1[4], n1[5], n1[6], n1[7]);
  wr2[0] = make_float4(n2[0], n2[1], n2[2], n2[3]);
  wr2[1] = make_float4(n2[4], n2[5], n2[6], n2[7]);

  float* g1 = agg + (size_t)v1 * CDIM;
  float* g2 = agg + (size_t)v2 * CDIM;
#pragma unroll
  for (int j = 0; j < CDIM; ++j) atomicAdd(&g1[j], n1[j]);
#pragma unroll
  for (int j = 0; j < CDIM; ++j) atomicAdd(&g2[j], n2[j]);
}

// ---------------- per-node renormalize (and re-zero agg) -------------------
__global__ __launch_bounds__(256) void k_node(
    const float* __restrict__ log_b0, float* __restrict__ agg,
    const float* __restrict__ scaling, float* __restrict__ log_b, int N) {
  int n = blockIdx.x * blockDim.x + threadIdx.x;
  if (n >= N) return;
  const float sc = scaling[n];
  size_t base = (size_t)n * CDIM;
  float v[CDIM], mx = -INFINITY;
#pragma unroll
  for (int j = 0; j < CDIM; ++j) {
    float g = agg[base + j];
    agg[base + j] = 0.0f;            // reset for next iteration
    // fwd: log_b0 + agg + (scaling-1)*agg == log_b0 + scaling*agg
    v[j] = log_b0[base + j] + sc * g;
    mx = fmaxf(mx, v[j]);
  }
  float s = 0.0f;
#pragma unroll
  for (int j = 0; j < CDIM; ++j) s += __expf(v[j] - mx);
  float lse = mx + __logf(s);
#pragma unroll
  for (int j = 0; j < CDIM; ++j) log_b[base + j] = v[j] - lse;
}

// ---------------------------------------------------------------------------
extern "C" void kernel_launch(void* const* d_in, const int* in_sizes, int n_in,
                              void* d_out, int out_size, void* d_ws,
                              size_t ws_size, hipStream_t stream) {
  // setup_inputs order: x, edge_index, rv, edge_weight, scaling, K, W, b, param
  const float* x       = (const float*)d_in[0];
  const int*   eidx    = (const int*)d_in[1];
  const float* ew      = (const float*)d_in[3];
  const float* scaling = (const float*)d_in[4];
  const float* W       = (const float*)d_in[6];
  const float* bias    = (const float*)d_in[7];
  const float* param   = (const float*)d_in[8];
  (void)d_in[2]; (void)d_in[5]; (void)n_in; (void)ws_size; (void)out_size;

  const int N  = in_sizes[4];       // nodes
  const int E  = in_sizes[3];       // directed edges (2*E2)
  const int E2 = E / 2;
  const int K  = 5;                 // K_ITERS (reference constant)

  const int* src = eidx;            // edge_index[0]
  const int* dst = eidx + E;        // edge_index[1]

  // workspace (floats): H[64] | Wpad[2048] | log_b0[N*8] | agg[N*8] | msg[E*8]
  float* ws      = (float*)d_ws;
  float* H       = ws;
  float* Wpad    = ws + 64;
  float* log_b0  = Wpad + (KDIM / 2) * 32;
  float* agg     = log_b0 + (size_t)N * CDIM;
  float* log_msg = agg + (size_t)N * CDIM;
  float* log_b   = (float*)d_out;

  k_logH<<<1, 64, 0, stream>>>(param, H);
  k_wpad<<<((KDIM / 2) * 32 + 255) / 256, 256, 0, stream>>>(W, Wpad);

  const int tiles = (N + 15) / 16;
  k_gemm_logsoftmax<<<(tiles + 7) / 8, 256, 0, stream>>>(x, Wpad, bias,
                                                         log_b0, log_b, N);

  const size_t n_msg = (size_t)E * CDIM;
  const size_t n_agg = (size_t)N * CDIM;
  k_init<<<(int)((n_msg + 255) / 256), 256, 0, stream>>>(log_msg, agg, n_msg,
                                                         n_agg);

  for (int it = 0; it < K; ++it) {
    k_edge<<<(E2 + 255) / 256, 256, 0, stream>>>(src, dst, ew, log_b, log_msg,
                                                 agg, H, E2);
    k_node<<<(N + 255) / 256, 256, 0, stream>>>(log_b0, agg, scaling, log_b,
                                                N);
  }
}